// VGAEModel_87978110091561
// MI455X (gfx1250) — compile-verified
//
#include <hip/hip_runtime.h>

typedef __attribute__((ext_vector_type(2))) float v2f;
typedef __attribute__((ext_vector_type(8))) float v8f;

// ---------------- elementwise / degree kernels ----------------

__global__ void zero_f32(float* __restrict__ p, long long n) {
  long long i = (long long)blockIdx.x * blockDim.x + threadIdx.x;
  long long stride = (long long)gridDim.x * blockDim.x;
  for (; i < n; i += stride) p[i] = 0.0f;
}

__global__ void degree_kernel(const long long* __restrict__ dst,
                              float* __restrict__ deg, int E) {
  int e = blockIdx.x * blockDim.x + threadIdx.x;
  if (e < E) {
    unsafeAtomicAdd(&deg[dst[e]], 1.0f);
  }
}

__global__ void dinv_kernel(float* __restrict__ deg, int n) {
  int i = blockIdx.x * blockDim.x + threadIdx.x;
  if (i < n) deg[i] = rsqrtf(deg[i] + 1.0f);
}

// agg[node][f] = Hp[node][f] * dinv[node]^2 + bias[f]
// launch: grid = N blocks, block = F threads
__global__ void init_agg(const float* __restrict__ Hp,
                         const float* __restrict__ dinv,
                         const float* __restrict__ bias,
                         float* __restrict__ agg, int F) {
  int node = blockIdx.x;
  int f = threadIdx.x;
  float di = dinv[node];
  size_t i = (size_t)node * F + f;
  agg[i] = Hp[i] * (di * di) + bias[f];
}

__global__ void relu_inplace(float* __restrict__ p, long long n) {
  long long i = (long long)blockIdx.x * blockDim.x + threadIdx.x;
  if (i < n) p[i] = fmaxf(p[i], 0.0f);
}

// agg[dst] += Hp[src] * (dinv[src]*dinv[dst]); one thread handles 4 features.
// Adjacent lanes -> adjacent feature chunks of the SAME edge (coalesced gather).
__global__ void scatter_edges(const float* __restrict__ Hp,
                              const long long* __restrict__ src,
                              const long long* __restrict__ dst,
                              const float* __restrict__ dinv,
                              float* __restrict__ agg,
                              int E, int logChunks, int F) {
  unsigned tid = blockIdx.x * blockDim.x + threadIdx.x;
  unsigned total = (unsigned)E << logChunks;
  if (tid >= total) return;
  unsigned e = tid >> logChunks;
  unsigned c = tid & ((1u << logChunks) - 1u);
  long long s = src[e];
  long long d = dst[e];
  float w = dinv[s] * dinv[d];
  const float4* hs = (const float4*)(Hp + (size_t)s * F);
  float4 v = hs[c];
  float* ad = agg + (size_t)d * F + (size_t)c * 4;
  unsafeAtomicAdd(ad + 0, v.x * w);
  unsafeAtomicAdd(ad + 1, v.y * w);
  unsafeAtomicAdd(ad + 2, v.z * w);
  unsafeAtomicAdd(ad + 3, v.w * w);
}

// ---------------- WMMA fp32 GEMM: H[nrows x Fout] = X[nrows x K] * W[K x Fout]
// One wave (32 lanes) per 16x16 output tile using V_WMMA_F32_16X16X4_F32.
// blockDim = (32, Fout/16); blockIdx.x = row tile.
// 32-bit operand layouts (ISA 7.12.2):
//   A 16x4 : a[v] = X[m][k0 + 2*half + v]     (m = lane&15, half = lane>>4)
//   B 4x16 : b[v] = W[k0 + 2*half + v][n]     (n = lane&15)
//   D 16x16: d[v] = H[16*tileM + v + 8*half][n]
__global__ void gemm_wmma_f32(const float* __restrict__ X,
                              const float* __restrict__ W,
                              float* __restrict__ H,
                              int nrows, int K, int Fout) {
  const int lane = threadIdx.x;
  const int half = lane >> 4;
  const int lm = lane & 15;
  const int tileM = blockIdx.x;
  const int tileN = threadIdx.y;

  int row = tileM * 16 + lm;
  int rclamp = row < nrows ? row : (nrows - 1);   // uniform-safe clamp for loads
  const float* xrow = X + (size_t)rclamp * K;
  const int coln = tileN * 16 + lm;

  v8f acc = {};
#pragma unroll 8
  for (int k0 = 0; k0 < K; k0 += 4) {
    int ka = k0 + 2 * half;
    v2f a;
    a.x = xrow[ka];
    a.y = xrow[ka + 1];
    v2f b;
    b.x = W[(size_t)ka * Fout + coln];
    b.y = W[(size_t)(ka + 1) * Fout + coln];
    // (neg_a, A, neg_b, B, c_mod, C, reuse_a, reuse_b)
    acc = __builtin_amdgcn_wmma_f32_16x16x4_f32(false, a, false, b,
                                                (short)0, acc, false, false);
  }

#pragma unroll
  for (int v = 0; v < 8; ++v) {
    int r = tileM * 16 + v + 8 * half;
    if (r < nrows) H[(size_t)r * Fout + coln] = acc[v];
  }
}

// ---------------- host launcher ----------------

static inline int ilog2i(int x) { int l = 0; while ((1 << l) < x) ++l; return l; }

extern "C" void kernel_launch(void* const* d_in, const int* in_sizes, int n_in,
                              void* d_out, int out_size, void* d_ws, size_t ws_size,
                              hipStream_t stream) {
  const float*      x   = (const float*)d_in[0];
  const long long*  ei  = (const long long*)d_in[1];   // int64 [2, E]
  const float*      W1  = (const float*)d_in[2];
  const float*      b1  = (const float*)d_in[3];
  const float*      Wmu = (const float*)d_in[4];
  const float*      bmu = (const float*)d_in[5];
  const float*      Wls = (const float*)d_in[6];
  const float*      bls = (const float*)d_in[7];

  const int IN_C  = 128;
  const int N     = in_sizes[0] / IN_C;          // 50000
  const int E     = in_sizes[1] / 2;             // 800000
  const int HID   = in_sizes[2] / IN_C;          // 128
  const int OUT_C = in_sizes[4] / HID;           // 64

  const long long* srcIdx = ei;
  const long long* dstIdx = ei + E;

  // carve workspace (floats, 256-elt aligned)
  float* ws = (float*)d_ws;
  size_t off = 0;
  auto carve = [&](size_t nf) {
    float* p = ws + off;
    off += (nf + 255) & ~(size_t)255;
    return p;
  };
  float* dinv = carve((size_t)N);
  float* H1p  = carve((size_t)N * HID);
  float* agg1 = carve((size_t)N * HID);
  float* Hmup = carve((size_t)N * OUT_C);
  float* Hlsp = carve((size_t)N * OUT_C);
  (void)ws_size;

  float* mu = (float*)d_out;
  float* ls = mu + (size_t)N * OUT_C;

  // 1) symmetric normalization: deg -> dinv = rsqrt(deg+1)
  zero_f32<<<256, 256, 0, stream>>>(dinv, (long long)N);
  degree_kernel<<<(E + 255) / 256, 256, 0, stream>>>(dstIdx, dinv, E);
  dinv_kernel<<<(N + 255) / 256, 256, 0, stream>>>(dinv, N);

  // 2) layer 1: H1p = x @ W1 ; agg1 = H1p*dinv^2 + b1 ; scatter ; relu
  {
    dim3 blk(32, HID / 16);
    gemm_wmma_f32<<<(N + 15) / 16, blk, 0, stream>>>(x, W1, H1p, N, IN_C, HID);
  }
  init_agg<<<N, HID, 0, stream>>>(H1p, dinv, b1, agg1, HID);
  {
    int lc = ilog2i(HID / 4);
    unsigned total = (unsigned)E << lc;
    scatter_edges<<<(total + 255) / 256, 256, 0, stream>>>(
        H1p, srcIdx, dstIdx, dinv, agg1, E, lc, HID);
  }
  relu_inplace<<<((long long)N * HID + 255) / 256, 256, 0, stream>>>(
      agg1, (long long)N * HID);

  // 3) layer 2 (mu and logstd), aggregating directly into d_out
  {
    dim3 blk(32, OUT_C / 16);
    gemm_wmma_f32<<<(N + 15) / 16, blk, 0, stream>>>(agg1, Wmu, Hmup, N, HID, OUT_C);
    gemm_wmma_f32<<<(N + 15) / 16, blk, 0, stream>>>(agg1, Wls, Hlsp, N, HID, OUT_C);
  }
  init_agg<<<N, OUT_C, 0, stream>>>(Hmup, dinv, bmu, mu, OUT_C);
  init_agg<<<N, OUT_C, 0, stream>>>(Hlsp, dinv, bls, ls, OUT_C);
  {
    int lc = ilog2i(OUT_C / 4);
    unsigned total = (unsigned)E << lc;
    scatter_edges<<<(total + 255) / 256, 256, 0, stream>>>(
        Hmup, srcIdx, dstIdx, dinv, mu, E, lc, OUT_C);
    scatter_edges<<<(total + 255) / 256, 256, 0, stream>>>(
        Hlsp, srcIdx, dstIdx, dinv, ls, E, lc, OUT_C);
  }
}